// RegularizedFMNet_22368189678476
// MI455X (gfx1250) — compile-verified
//
#include <hip/hip_runtime.h>
#include <hip/hip_bf16.h>

typedef __attribute__((ext_vector_type(2))) float v2f;
typedef __attribute__((ext_vector_type(8))) float v8f;

#define KDIM 128
#define PDIM 256
#define NDIM 100000
#define LAMBDA 0.001f
#define NCHUNKS 64
#define CHUNK ((NDIM + NCHUNKS - 1) / NCHUNKS)   // 1563
#define KP (KDIM * PDIM)                          // 32768

// ---------------------------------------------------------------------------
// Kernel 1: split-K partial GEMM.  grid = (NCHUNKS, 2), block = 512 (16 waves)
// sel = blockIdx.y: 0 -> A partial from (evecs_x, feat_x), 1 -> B partial.
// Each WG computes the full 128x256 partial product for its N-chunk with
// v_wmma_f32_16x16x4_f32.  LDS staging is double-buffered through registers:
// the next 32-row slab's global loads are issued before the current slab's
// 64 WMMAs, so HBM latency overlaps matrix compute.
// ---------------------------------------------------------------------------
__global__ __launch_bounds__(512) void fm_gemm_partial(
    const float* __restrict__ evx, const float* __restrict__ evy,
    const float* __restrict__ fx,  const float* __restrict__ fy,
    float* __restrict__ partials)
{
    const int sel   = blockIdx.y;
    const int chunk = blockIdx.x;
    const float* __restrict__ ev   = sel ? evy : evx;
    const float* __restrict__ feat = sel ? fy  : fx;

    __shared__ float evBlk[KDIM][33];     // 128 x 32 N-slab, padded (odd stride)
    __shared__ float ftBlk[32][260];      // 32 N-slab x 256 cols, padded

    const int tid  = threadIdx.x;
    const int lane = tid & 31;
    const int wave = tid >> 5;            // 0..15
    const int rowBand = (wave >> 1) * 16; // 16-row band of K
    const int wcol    = (wave & 1) * 128; // column half of P

    // per-thread staging assignments (fixed: 8 evec elems, 16 feat elems)
    // evecs: idx = tid + 512*q, q=0..7 -> row idx>>5, col idx&31
    // feat:  idx = tid + 512*q, q=0..15 -> row idx>>8, col idx&255
    v8f acc[8];
    #pragma unroll
    for (int t = 0; t < 8; ++t) acc[t] = (v8f)0.0f;

    const int nStart = chunk * CHUNK;
    const int nEnd   = min(nStart + CHUNK, NDIM);

    // ---- preload first slab straight into LDS ----
    {
        const int blk = nStart;
        #pragma unroll
        for (int q = 0; q < 8; ++q) {
            int idx = tid + 512 * q;
            int r = idx >> 5, c = idx & 31;
            int n = blk + c;
            evBlk[r][c] = (n < nEnd) ? ev[(long)r * NDIM + n] : 0.0f;
        }
        #pragma unroll
        for (int q = 0; q < 16; ++q) {
            int idx = tid + 512 * q;
            int r = idx >> 8, c = idx & 255;
            int n = blk + r;
            ftBlk[r][c] = (n < nEnd) ? feat[(long)n * PDIM + c] : 0.0f;
        }
    }
    __syncthreads();

    const int m  = lane & 15;
    const int kp = (lane >> 4) << 1;      // 0 or 2

    for (int blk = nStart; blk < nEnd; blk += 32) {
        const int nxt = blk + 32;
        const bool more = (nxt < nEnd);   // uniform across the workgroup

        // ---- issue next slab's global loads into registers (in flight
        //      while the WMMA burst below executes) ----
        float evr[8];
        float ftr[16];
        if (more) {
            #pragma unroll
            for (int q = 0; q < 8; ++q) {
                int idx = tid + 512 * q;
                int r = idx >> 5, c = idx & 31;
                int n = nxt + c;
                evr[q] = (n < nEnd) ? ev[(long)r * NDIM + n] : 0.0f;
            }
            #pragma unroll
            for (int q = 0; q < 16; ++q) {
                int idx = tid + 512 * q;
                int r = idx >> 8, c = idx & 255;
                int n = nxt + r;
                ftr[q] = (n < nEnd) ? feat[(long)n * PDIM + c] : 0.0f;
            }
            // cacheline-granular prefetch two slabs ahead
            if (nxt + 32 < nEnd) {
                long base = (long)(nxt + 32) * PDIM;
                __builtin_prefetch(&feat[base + (long)tid * 64], 0, 0);
            }
        }

        // ---- 64 WMMAs on the current LDS slab ----
        #pragma unroll
        for (int kk = 0; kk < 8; ++kk) {  // 8 k-steps of K=4 cover the 32-slab
            const int nOff = kk * 4;
            v2f a;
            a.x = evBlk[rowBand + m][nOff + kp];
            a.y = evBlk[rowBand + m][nOff + kp + 1];
            #pragma unroll
            for (int t = 0; t < 8; ++t) {
                const int col = wcol + t * 16 + m;
                v2f b;
                b.x = ftBlk[nOff + kp][col];
                b.y = ftBlk[nOff + kp + 1][col];
                acc[t] = __builtin_amdgcn_wmma_f32_16x16x4_f32(
                    false, a, false, b, (short)0, acc[t], false, false);
            }
        }
        __syncthreads();

        // ---- dump staged registers into LDS for the next iteration ----
        if (more) {
            #pragma unroll
            for (int q = 0; q < 8; ++q) {
                int idx = tid + 512 * q;
                evBlk[idx >> 5][idx & 31] = evr[q];
            }
            #pragma unroll
            for (int q = 0; q < 16; ++q) {
                int idx = tid + 512 * q;
                ftBlk[idx >> 8][idx & 255] = ftr[q];
            }
            __syncthreads();
        }
    }

    // write 128x256 partial (deterministic per-chunk slot)
    float* out = partials + (size_t)sel * NCHUNKS * KP + (size_t)chunk * KP;
    const int mrow = rowBand + ((lane >> 4) << 3); // +8 for lanes 16..31
    #pragma unroll
    for (int t = 0; t < 8; ++t) {
        const int col = wcol + t * 16 + (lane & 15);
        #pragma unroll
        for (int r = 0; r < 8; ++r)
            out[(mrow + r) * PDIM + col] = acc[t][r];
    }
}

// ---------------------------------------------------------------------------
// Kernel 2: deterministic reduction of split-K partials -> A, B (128x256 each)
// ---------------------------------------------------------------------------
__global__ void fm_reduce(const float* __restrict__ partials,
                          float* __restrict__ Amat, float* __restrict__ Bmat)
{
    int idx = blockIdx.x * blockDim.x + threadIdx.x;
    if (idx >= KP) return;
    float sa = 0.0f, sb = 0.0f;
    for (int c = 0; c < NCHUNKS; ++c) {
        sa += partials[(size_t)c * KP + idx];
        sb += partials[(size_t)(NCHUNKS + c) * KP + idx];
    }
    Amat[idx] = sa;
    Bmat[idx] = sb;
}

// ---------------------------------------------------------------------------
// Kernel 3: mask D (128x128).  evals are sorted ascending -> max is last elem.
// gamma = 0.5 -> sqrt.
// ---------------------------------------------------------------------------
__global__ void fm_mask(const float* __restrict__ evx,
                        const float* __restrict__ evy,
                        float* __restrict__ D)
{
    int idx = blockIdx.x * blockDim.x + threadIdx.x;
    if (idx >= KDIM * KDIM) return;
    int i = idx >> 7;      // row: indexes evals_y
    int j = idx & 127;     // col: indexes evals_x
    float s  = fmaxf(evx[KDIM - 1], evy[KDIM - 1]);
    float g1 = sqrtf(evx[j] / s);
    float g2 = sqrtf(evy[i] / s);
    float d1 = 1.0f / (g1 * g1 + 1.0f);
    float d2 = 1.0f / (g2 * g2 + 1.0f);
    float re = g2 * d2 - g1 * d1;
    float im = d2 - d1;
    D[idx] = re * re + im * im;
}

// ---------------------------------------------------------------------------
// Kernel 4: Gram matrices via WMMA.  grid = 2 (block0: AAt = A·Aᵀ,
// block1: BAt = B·Aᵀ), block = 256 (8 waves).  Since Bmat(k=p, n=j) = R[j][p],
// the B-fragment is the same row-gather pattern as the A-fragment.
// ---------------------------------------------------------------------------
__global__ __launch_bounds__(256) void fm_gram(
    const float* __restrict__ Amat, const float* __restrict__ Bmat,
    float* __restrict__ AAt, float* __restrict__ BAt)
{
    const float* __restrict__ L = blockIdx.x ? Bmat : Amat;
    const float* __restrict__ R = Amat;
    float* __restrict__ out     = blockIdx.x ? BAt : AAt;

    __shared__ float blkL[KDIM][33];
    __shared__ float blkR[KDIM][33];

    const int tid  = threadIdx.x;
    const int lane = tid & 31;
    const int wave = tid >> 5;        // 0..7 -> 16-row band
    const int iBase = wave * 16;

    v8f acc[8];
    #pragma unroll
    for (int t = 0; t < 8; ++t) acc[t] = (v8f)0.0f;

    for (int p0 = 0; p0 < PDIM; p0 += 32) {
        for (int idx = tid; idx < KDIM * 32; idx += 256) {
            int r = idx >> 5, c = idx & 31;
            blkL[r][c] = L[r * PDIM + p0 + c];
            blkR[r][c] = R[r * PDIM + p0 + c];
        }
        __syncthreads();

        const int m  = lane & 15;
        const int kp = (lane >> 4) << 1;
        #pragma unroll
        for (int kk = 0; kk < 8; ++kk) {
            const int pOff = kk * 4;
            v2f a;
            a.x = blkL[iBase + m][pOff + kp];
            a.y = blkL[iBase + m][pOff + kp + 1];
            #pragma unroll
            for (int t = 0; t < 8; ++t) {
                v2f b;
                b.x = blkR[t * 16 + m][pOff + kp];
                b.y = blkR[t * 16 + m][pOff + kp + 1];
                acc[t] = __builtin_amdgcn_wmma_f32_16x16x4_f32(
                    false, a, false, b, (short)0, acc[t], false, false);
            }
        }
        __syncthreads();
    }

    const int mrow = iBase + ((lane >> 4) << 3);
    #pragma unroll
    for (int t = 0; t < 8; ++t) {
        const int j = t * 16 + (lane & 15);
        #pragma unroll
        for (int r = 0; r < 8; ++r)
            out[(mrow + r) * KDIM + j] = acc[t][r];
    }
}

// ---------------------------------------------------------------------------
// Kernel 5: 128 independent SPD solves, one workgroup each.
// M_i = AAt + LAMBDA*diag(D[i,:]),  M_i x = BAt[i,:].
// Gaussian elimination (no pivoting; SPD) in dynamic LDS, stride 129
// (odd stride -> conflict-free across the 64 banks). RHS lives in column 128.
// ---------------------------------------------------------------------------
__global__ __launch_bounds__(128) void fm_solve(
    const float* __restrict__ AAt, const float* __restrict__ BAt,
    const float* __restrict__ D,   float* __restrict__ Cout)
{
    extern __shared__ float Msh[];     // 128 x 129
    const int i   = blockIdx.x;        // system index (row of D / BAt)
    const int tid = threadIdx.x;

    for (int idx = tid; idx < KDIM * 129; idx += 128) {
        int r = idx / 129, c = idx - r * 129;
        float v;
        if (c < KDIM) {
            v = AAt[r * KDIM + c];
            if (c == r) v += LAMBDA * D[i * KDIM + r];
        } else {
            v = BAt[i * KDIM + r];
        }
        Msh[idx] = v;
    }

    // forward elimination: thread r eliminates its own row at each pivot step
    for (int c = 0; c < KDIM - 1; ++c) {
        __syncthreads();
        int r = tid;
        if (r > c) {
            float f = Msh[r * 129 + c] / Msh[c * 129 + c];
            for (int j = c + 1; j <= KDIM; ++j)
                Msh[r * 129 + j] -= f * Msh[c * 129 + j];
        }
    }

    // back substitution (x accumulates in column 128)
    for (int c = KDIM - 1; c >= 0; --c) {
        __syncthreads();
        if (tid == c)
            Msh[c * 129 + KDIM] /= Msh[c * 129 + c];
        __syncthreads();
        if (tid < c)
            Msh[tid * 129 + KDIM] -= Msh[tid * 129 + c] * Msh[c * 129 + KDIM];
    }
    __syncthreads();
    Cout[i * KDIM + tid] = Msh[tid * 129 + KDIM];
}

// ---------------------------------------------------------------------------
// Workspace layout (floats):
//   [0, 32768)          A
//   [32768, 65536)      B
//   [65536, 81920)      AAt
//   [81920, 98304)      BAt
//   [98304, 114688)     D
//   [114688, ...)       split-K partials: 2 * NCHUNKS * 32768 floats
// Total ~17.2 MB.
// ---------------------------------------------------------------------------
extern "C" void kernel_launch(void* const* d_in, const int* in_sizes, int n_in,
                              void* d_out, int out_size, void* d_ws, size_t ws_size,
                              hipStream_t stream)
{
    const float* feat_x = (const float*)d_in[0];
    const float* feat_y = (const float*)d_in[1];
    const float* evals_x = (const float*)d_in[2];
    const float* evals_y = (const float*)d_in[3];
    const float* evecs_x = (const float*)d_in[4];
    const float* evecs_y = (const float*)d_in[5];

    float* wsf   = (float*)d_ws;
    float* Amat  = wsf;
    float* Bmat  = wsf + 32768;
    float* AAt   = wsf + 65536;
    float* BAt   = wsf + 81920;
    float* Dmask = wsf + 98304;
    float* part  = wsf + 114688;
    float* Cout  = (float*)d_out;

    // 1) split-K GEMM partials (both tensors)
    fm_gemm_partial<<<dim3(NCHUNKS, 2), 512, 0, stream>>>(
        evecs_x, evecs_y, feat_x, feat_y, part);

    // 2) deterministic reduction -> A, B
    fm_reduce<<<(KP + 255) / 256, 256, 0, stream>>>(part, Amat, Bmat);

    // 3) mask D
    fm_mask<<<(KDIM * KDIM + 255) / 256, 256, 0, stream>>>(evals_x, evals_y, Dmask);

    // 4) Gram matrices AAt, BAt
    fm_gram<<<2, 256, 0, stream>>>(Amat, Bmat, AAt, BAt);

    // 5) 128 regularized solves -> C
    fm_solve<<<KDIM, KDIM, KDIM * 129 * sizeof(float), stream>>>(
        AAt, BAt, Dmask, Cout);
}